// SoftGumbelQuantizedMaskLayer_21277267984950
// MI455X (gfx1250) — compile-verified
//
#include <hip/hip_runtime.h>

// Fused SoftGumbelQuantizedMaskLayer: mask quantization + phase shift + aperture
// + complex field modulation for MI455X (gfx1250, wave32).
//
// Pure streaming elementwise op => memory-bound. Optimizations:
//  * one thread = 4 pixels => all VMEM is 128-bit (global_load_b128/store_b128)
//  * non-temporal cache policy on single-use streams (don't churn 192MB L2)
//  * aperture culling: outside |x|,|y| <= 40mm the result is exactly 0 =>
//    skip ALL input loads there (live region is only ~15% of pixels)
//  * mask/height computed once per pixel, reused across 4 wavelengths
//  * hardware transcendentals: v_exp_f32 / v_sin_f32 / v_cos_f32 / v_rcp_f32

typedef float v4f __attribute__((ext_vector_type(4)));

#define HH 2048
#define WW 2048
#define NWL 4
#define HWSZ (HH * WW)

__device__ __forceinline__ float fast_rcp(float x) {
    return __builtin_amdgcn_rcpf(x);
}

__device__ __forceinline__ float fast_sigmoid(float z) {
    // 1/(1+exp(-z)) : v_exp_f32 + v_rcp_f32
    return fast_rcp(1.0f + __expf(-z));
}

__device__ __forceinline__ float mod4(float x) {
    // floor-mod by 4 (matches jnp.mod for divisor 4): result in [0,4)
    return x - 4.0f * floorf(x * 0.25f);
}

__global__ __launch_bounds__(256) void soft_gumbel_mask_modulate(
    const float* __restrict__ init_mask,   // [H*W]
    const float* __restrict__ lut,         // [2]
    const float* __restrict__ field_real,  // [NWL*H*W]
    const float* __restrict__ field_imag,  // [NWL*H*W]
    const float* __restrict__ wavelengths, // [NWL]
    const float* __restrict__ gumbel,      // [2*H*W]
    const float* __restrict__ epsilon,     // [1]
    const float* __restrict__ tand,        // [1]
    float* __restrict__ out)               // [2*NWL*H*W] (real block, imag block)
{
    constexpr float PI        = 3.14159265358979323846f;
    constexpr float BASE_T    = 2.0e-3f;    // 2 mm
    constexpr float DXP       = 1.0e-4f;    // pixel pitch
    constexpr float HALF_RECT = 40.0e-3f;   // 80 mm aperture / 2

    // ---- uniform scalars: lowered to s_load, shared by the whole wave ----
    const float eps  = epsilon[0];
    const float td   = tand[0];
    const float sq   = sqrtf(eps);
    const float lutA = lut[0];
    const float lutB = lut[1];

    float lam[NWL];
#pragma unroll
    for (int w = 0; w < NWL; ++w) lam[w] = wavelengths[w];
    float lam_min = fminf(fminf(lam[0], lam[1]), fminf(lam[2], lam[3]));
    const float hcoef = lam_min * 0.5f / (sq - 1.0f); // height = mask * hcoef

    float acoef[NWL], bcoef[NWL];
#pragma unroll
    for (int w = 0; w < NWL; ++w) {
        float k  = 2.0f * PI / lam[w];
        acoef[w] = -0.5f * k * td * sq;   // loss  = exp(acoef * h)
        bcoef[w] = -k * (sq - 1.0f);      // angle = bcoef * h
    }

    // ---- pixel coordinates: thread t covers pixels [x0, x0+3] of row y ----
    const int t  = blockIdx.x * blockDim.x + threadIdx.x;
    const int x0 = (t & (WW / 4 - 1)) * 4;   // WW/4 = 512
    const int y  = t >> 9;
    const int base = y * WW + x0;

    // ---- rect aperture (linspace endpoints match jnp.linspace) ----
    const float ystep = (DXP * HH) / (float)(HH - 1);
    const float Y     = -DXP * HH * 0.5f + (float)y * ystep;
    const bool  row_in = (fabsf(Y) <= HALF_RECT);
    const float xstep = (DXP * WW) / (float)(WW - 1);
    const float xbase = -DXP * WW * 0.5f;

    float apx[4];
    bool  any_in = false;
#pragma unroll
    for (int i = 0; i < 4; ++i) {
        float X  = xbase + (float)(x0 + i) * xstep;
        bool  in = row_in && (fabsf(X) <= HALF_RECT);
        apx[i]   = in ? 1.0f : 0.0f;
        any_in  |= in;
    }

    if (!any_in) {
        // Entirely outside aperture: output is exactly zero; no input loads.
        v4f z = (v4f)0.0f;
#pragma unroll
        for (int w = 0; w < NWL; ++w) {
            __builtin_nontemporal_store(z, (v4f*)(out + (size_t)w * HWSZ + base));
            __builtin_nontemporal_store(z, (v4f*)(out + (size_t)(NWL + w) * HWSZ + base));
        }
        return;
    }

    // ---- per-pixel soft-quantized mask -> height (single-use: NT loads) ----
    v4f m  = __builtin_nontemporal_load((const v4f*)(init_mask + base));
    v4f g0 = __builtin_nontemporal_load((const v4f*)(gumbel + base));
    v4f g1 = __builtin_nontemporal_load((const v4f*)(gumbel + HWSZ + base));

    float hv[4];
#pragma unroll
    for (int i = 0; i < 4; ++i) {
        float wrapped = mod4(m[i] + 2.0f) - 2.0f;
        float d0 = (mod4(wrapped - lutA + 2.0f) - 2.0f) * 0.5f;
        float d1 = (mod4(wrapped - lutB + 2.0f) - 2.0f) * 0.5f;
        float s0 = fast_sigmoid(5.0f * d0);
        float s1 = fast_sigmoid(5.0f * d1);
        float sc0 = 4.0f * s0 * (1.0f - s0);
        float sc1 = 4.0f * s1 * (1.0f - s1);
        // 2-way softmax((score+g)/tau) with tau=2 => sigmoid of the logit gap
        float p1   = fast_sigmoid((sc1 + g1[i]) * 0.5f - (sc0 + g0[i]) * 0.5f);
        float mask = lutA + p1 * (lutB - lutA);
        hv[i] = mask * hcoef + BASE_T;           // height + base thickness
    }

    // ---- per-wavelength complex modulation ----
#pragma unroll
    for (int w = 0; w < NWL; ++w) {
        const size_t foff = (size_t)w * HWSZ + base;
        v4f fr = __builtin_nontemporal_load((const v4f*)(field_real + foff));
        v4f fi = __builtin_nontemporal_load((const v4f*)(field_imag + foff));
        v4f ore, oim;
#pragma unroll
        for (int i = 0; i < 4; ++i) {
            float loss = __expf(acoef[w] * hv[i]) * apx[i];
            float ang  = bcoef[w] * hv[i];
            float s, c;
            __sincosf(ang, &s, &c);
            ore[i] = loss * (fr[i] * c - fi[i] * s);
            oim[i] = loss * (fr[i] * s + fi[i] * c);
        }
        __builtin_nontemporal_store(ore, (v4f*)(out + (size_t)w * HWSZ + base));
        __builtin_nontemporal_store(oim, (v4f*)(out + (size_t)(NWL + w) * HWSZ + base));
    }
}

extern "C" void kernel_launch(void* const* d_in, const int* in_sizes, int n_in,
                              void* d_out, int out_size, void* d_ws, size_t ws_size,
                              hipStream_t stream) {
    (void)in_sizes; (void)n_in; (void)out_size; (void)d_ws; (void)ws_size;
    const float* init_mask   = (const float*)d_in[0];
    const float* lut         = (const float*)d_in[1];
    const float* field_real  = (const float*)d_in[2];
    const float* field_imag  = (const float*)d_in[3];
    const float* wavelengths = (const float*)d_in[4];
    const float* gumbel      = (const float*)d_in[5];
    const float* epsilon     = (const float*)d_in[6];
    const float* tand        = (const float*)d_in[7];
    float* out = (float*)d_out;

    const int nthreads = HWSZ / 4;            // 1,048,576 threads (4 px each)
    const int block = 256;                    // 8 wave32 per workgroup
    const int grid  = nthreads / block;       // 4096 blocks
    soft_gumbel_mask_modulate<<<grid, block, 0, stream>>>(
        init_mask, lut, field_real, field_imag, wavelengths, gumbel,
        epsilon, tand, out);
}